// TestModel_84645215469631
// MI455X (gfx1250) — compile-verified
//
#include <hip/hip_runtime.h>

// CDNA5 (gfx1250, wave32) kernel:
// 12-qubit product state + CNOT-chain (== Gray-code reindex) + Z-parity observable.
// Per (b,nb) pair: out[j_hi*64+j_lo] = Chi[gray6(j_hi)] * Clo[gray6(j_lo) ^ 32*lsb(j_hi)]
// -> rank-2 64x64 complex outer product, computed with V_WMMA_F32_16X16X4_F32 (K=4).
// O = prod_{q odd} cos(theta_q) (analytic collapse of sum zd*|psi|^2).

typedef __attribute__((ext_vector_type(2))) float v2f;
typedef __attribute__((ext_vector_type(8))) float v8f;

#define NQ_ 12
#define DIM_ 4096
#define NPAIR 1024   // B(64) * NB(16)

__global__ __launch_bounds__(128) void qstate_wmma_kernel(const float* __restrict__ x,
                                                          float* __restrict__ out) {
    __shared__ float s_c[NQ_], s_s[NQ_];
    __shared__ float s_ar[64], s_ai[64];  // Chi (qubits 0..5), with (-i)^popcount phase folded in
    __shared__ float s_br[64], s_bi[64];  // Clo (qubits 6..11)

    const int tid  = threadIdx.x;
    const int pair = blockIdx.x;

    // ---- Phase 1: 12 half-angle sin/cos per pair ----
    if (tid < NQ_) {
        const float th = x[pair * NQ_ + tid];
        float sn, cs;
        __sincosf(0.5f * th, &sn, &cs);
        s_c[tid] = cs;
        s_s[tid] = sn;
    }
    __syncthreads();

    // ---- Phase 2: 64-entry complex half-state tables (threads 0..63 -> Chi, 64..127 -> Clo) ----
    {
        const int  h  = tid & 63;
        const bool lo = tid >= 64;
        float mag = 1.0f;
#pragma unroll
        for (int p = 0; p < 6; ++p) {
            const int q = (lo ? 11 : 5) - p;          // qubit index for bit position p of h
            mag *= ((h >> p) & 1) ? s_s[q] : s_c[q];
        }
        const int pc = __popc(h) & 3;                  // (-i)^popcount
        float re = 0.0f, im = 0.0f;
        if      (pc == 0) re =  mag;
        else if (pc == 1) im = -mag;
        else if (pc == 2) re = -mag;
        else              im =  mag;
        if (lo) { s_br[h] = re; s_bi[h] = im; }
        else    { s_ar[h] = re; s_ai[h] = im; }
    }
    __syncthreads();

    // ---- Phase 3: 64x64 complex outer product via WMMA f32 16x16x4 ----
    // Wave w owns row-block w (tile_m = w), loops over 4 column tiles.
    const int lane = tid & 31;
    const int wave = tid >> 5;
    const int hi   = lane >> 4;     // half-wave: selects K={0,1} vs K={2,3} rows of A/B
    const int ln   = lane & 15;

    const int  tile_m = wave;                     // 0..3
    const int  am     = tile_m * 16 + ln;         // output row j_hi handled by this lane's A slot
    const int  gm     = (am ^ (am >> 1)) & 63;    // gray6(j_hi)
    const bool asel   = ((am & 1) == hi);         // even rows -> K0/K1, odd rows -> K2/K3
    v2f A;
    A.x = asel ? s_ar[gm] : 0.0f;                 // A[m][K0 or K2] = Re Chi
    A.y = asel ? s_ai[gm] : 0.0f;                 // A[m][K1 or K3] = Im Chi

    float* basep = out + (size_t)pair * (DIM_ * 2);

#pragma unroll
    for (int tn = 0; tn < 4; ++tn) {
        const int n  = tn * 16 + ln;                           // output col j_lo
        const int gn = ((n ^ (n >> 1)) & 63) ^ (hi << 5);      // gray6(j_lo) ^ 32*lsb(j_hi)
        const float br = s_br[gn], bi = s_bi[gn];
        v2f Bre, Bim;
        Bre.x = br;  Bre.y = -bi;   // Re(a*b) = ar*br - ai*bi
        Bim.x = bi;  Bim.y =  br;   // Im(a*b) = ar*bi + ai*br
        v8f zero = {};
        // 8 args: (neg_a, A, neg_b, B, c_mod, C, reuse_a, reuse_b)
        v8f Dre = __builtin_amdgcn_wmma_f32_16x16x4_f32(false, A, false, Bre, (short)0, zero, false, false);
        v8f Dim = __builtin_amdgcn_wmma_f32_16x16x4_f32(false, A, false, Bim, (short)0, zero, false, false);
        // D layout: VGPR r -> row (tile_m*16 + r + 8*hi), col n. Interleaved complex, 8B stores.
#pragma unroll
        for (int r = 0; r < 8; ++r) {
            const int mrow = tile_m * 16 + r + 8 * hi;
            v2f val;
            val.x = Dre[r];
            val.y = Dim[r];
            *(v2f*)(basep + ((size_t)mrow * 64 + n) * 2) = val;
        }
    }

    // ---- Phase 4: observable O = prod over odd qubits of cos(theta) (cos^2 - sin^2 of half angle) ----
    if (tid == 0) {
        float o = 1.0f;
#pragma unroll
        for (int q = 1; q < NQ_; q += 2) o *= (s_c[q] * s_c[q] - s_s[q] * s_s[q]);
        const size_t obase = (size_t)NPAIR * DIM_ * 2;
        out[obase + (size_t)pair * 2 + 0] = o;
        out[obase + (size_t)pair * 2 + 1] = 0.0f;
    }
}

extern "C" void kernel_launch(void* const* d_in, const int* in_sizes, int n_in,
                              void* d_out, int out_size, void* d_ws, size_t ws_size,
                              hipStream_t stream) {
    (void)in_sizes; (void)n_in; (void)d_ws; (void)ws_size; (void)out_size;
    const float* x = (const float*)d_in[0];        // (64,16,12,1) float32, pair-major
    float* out = (float*)d_out;                     // [1024*4096 complex | 1024 complex] as (re,im) floats
    hipLaunchKernelGGL(qstate_wmma_kernel, dim3(NPAIR), dim3(128), 0, stream, x, out);
}